// MSClassifyNet_35270271435011
// MI455X (gfx1250) — compile-verified
//
#include <hip/hip_runtime.h>
#include <hip/hip_bf16.h>

// ---------------------------------------------------------------------------
// MSClassifyNet — S=2048, batch 1. Chain of GEMVs + 2 big elementwise maps.
// GEMVs: V_WMMA_F32_16X16X4_F32 (f32-exact), K-split x32 for memory
// parallelism, partials reduced by a tiny epilogue kernel. ~370MB total HBM
// traffic -> ~16us roofline at 23.3 TB/s.
// ---------------------------------------------------------------------------

typedef __attribute__((ext_vector_type(2))) float v2f;
typedef __attribute__((ext_vector_type(4))) float v4f;
typedef __attribute__((ext_vector_type(8))) float v8f;

#define S2048 2048
#define KSPLIT 32

enum { ACT_NONE = 0, ACT_TANH = 1, ACT_RELU = 2, ACT_LAM = 3 };

__device__ __forceinline__ float sigmf(float v) { return 1.f / (1.f + expf(-v)); }

// ---------------------------------------------------------------------------
// Partial GEMV via WMMA f32 16x16x4.
//   part[blockIdx.y*FOUT + j] = sum_{i in K-chunk} x[i] * W[i, j]
// W row-major (fin x FOUT). One wave -> 16 consecutive output columns.
// A operand (16x4 f32): row M=0 carries x (lanes 0/16 = K0,1 / K2,3),
// zeroed elsewhere via multiply-mask (no EXEC divergence).
// B (4x16): lanes0-15 = row k+krow col n; second row at constant offset FOUT.
// ---------------------------------------------------------------------------
template <int FOUT>
__global__ __launch_bounds__(256) void gemv_part_k(
    const float* __restrict__ W, const float* __restrict__ x,
    float* __restrict__ part, int fin) {
  __shared__ float xs[192];                       // max chunk = 6144/32
  const int chunk = fin / KSPLIT;
  const int kbase = blockIdx.y * chunk;
  for (int t = threadIdx.x; t < chunk; t += 256) xs[t] = x[kbase + t];
  __syncthreads();
  const int lane = threadIdx.x & 31;
  const int wave = threadIdx.x >> 5;
  const int j0 = (blockIdx.x * 8 + wave) * 16;
  const int n = lane & 15;
  const int krow = (lane >> 4) * 2;               // 0 | 2
  const float am = (n == 0) ? 1.f : 0.f;          // A-row-0 mask
  v8f acc = {0.f, 0.f, 0.f, 0.f, 0.f, 0.f, 0.f, 0.f};
  const float* wp = W + (size_t)(kbase + krow) * FOUT + j0 + n;
  for (int k = 0; k < chunk; k += 4) {
    v2f a;
    a[0] = xs[k + krow] * am;
    a[1] = xs[k + krow + 1] * am;
    v2f b;
    b[0] = __builtin_nontemporal_load(wp);
    b[1] = __builtin_nontemporal_load(wp + FOUT);
    wp += 4 * FOUT;
    acc = __builtin_amdgcn_wmma_f32_16x16x4_f32(false, a, false, b, (short)0,
                                                acc, false, false);
  }
  if (lane < 16) part[blockIdx.y * FOUT + j0 + n] = acc[0];
}

// Batched partial GEMV: two vectors share one weight-matrix read
// (A rows M=0 and M=1). FOUT fixed 2048, fin fixed 2048 (attention stages).
__global__ __launch_bounds__(256) void gemv2_part_k(
    const float* __restrict__ W, const float* __restrict__ x0,
    const float* __restrict__ x1, float* __restrict__ part0,
    float* __restrict__ part1) {
  constexpr int FOUT = 2048;
  constexpr int chunk = 2048 / KSPLIT;            // 64
  __shared__ float xs[2 * chunk];
  const int kbase = blockIdx.y * chunk;
  if (threadIdx.x < chunk) xs[threadIdx.x] = x0[kbase + threadIdx.x];
  else if (threadIdx.x < 2 * chunk) xs[threadIdx.x] = x1[kbase + threadIdx.x - chunk];
  __syncthreads();
  const int lane = threadIdx.x & 31;
  const int wave = threadIdx.x >> 5;
  const int j0 = (blockIdx.x * 8 + wave) * 16;
  const int n = lane & 15;
  const int krow = (lane >> 4) * 2;
  const float am = (n < 2) ? 1.f : 0.f;
  const int xb = (n == 1) ? chunk : 0;
  v8f acc = {0.f, 0.f, 0.f, 0.f, 0.f, 0.f, 0.f, 0.f};
  const float* wp = W + (size_t)(kbase + krow) * FOUT + j0 + n;
  for (int k = 0; k < chunk; k += 4) {
    v2f a;
    a[0] = xs[xb + k + krow] * am;
    a[1] = xs[xb + k + krow + 1] * am;
    v2f b;
    b[0] = __builtin_nontemporal_load(wp);
    b[1] = __builtin_nontemporal_load(wp + FOUT);
    wp += 4 * FOUT;
    acc = __builtin_amdgcn_wmma_f32_16x16x4_f32(false, a, false, b, (short)0,
                                                acc, false, false);
  }
  if (lane < 16) {
    part0[blockIdx.y * FOUT + j0 + n] = acc[0];   // D row 0
    part1[blockIdx.y * FOUT + j0 + n] = acc[1];   // D row 1
  }
}

// Epilogue: out[j] = act( sum_s part[s*fout+j] + bias[j] )
template <int ACT, bool TWO_OUT>
__global__ __launch_bounds__(256) void reduce_act_k(
    const float* __restrict__ part, const float* __restrict__ bias,
    float* __restrict__ out, float* __restrict__ out2, int fout,
    const float* __restrict__ nw, const float* __restrict__ tempp) {
  const int j = blockIdx.x * 256 + threadIdx.x;
  if (j >= fout) return;
  float r = bias[j];
  for (int s = 0; s < KSPLIT; ++s) r += part[s * fout + j];
  if (ACT == ACT_TANH) {
    float u = tanhf(r);
    out[j] = u;
    if (TWO_OUT) out2[j] = tanhf(u);
  } else if (ACT == ACT_RELU) {
    out[j] = fmaxf(r, 0.f);
  } else if (ACT == ACT_LAM) {
    float c = sigmf(r);
    out[j] = sigmf(c * sigmf(nw[j]) / tempp[0]);
  } else {
    out[j] = r;
  }
}

// Epilogue for the batched GEMV (both plain affine).
__global__ __launch_bounds__(256) void reduce2_k(
    const float* __restrict__ part0, const float* __restrict__ part1,
    const float* __restrict__ bias, float* __restrict__ out0,
    float* __restrict__ out1) {
  const int fout = 2048;
  const int j = blockIdx.x * 256 + threadIdx.x;
  float r0 = bias[j], r1 = bias[j];
  for (int s = 0; s < KSPLIT; ++s) {
    r0 += part0[s * fout + j];
    r1 += part1[s * fout + j];
  }
  out0[j] = r0;
  out1[j] = r1;
}

// ---------------------------------------------------------------------------
// Elementwise / reduction kernels
// ---------------------------------------------------------------------------
__global__ __launch_bounds__(256) void gprep_k(const float* __restrict__ bsc,
                                               const float* __restrict__ bfc,
                                               float* __restrict__ gsc,
                                               float* __restrict__ gfc) {
  int i = blockIdx.x * 256 + threadIdx.x;         // grid covers 2048
  gsc[i] = sigmf(bsc[i]);
  gfc[i] = sigmf(bfc[i]);
}

// B_s = sc * g_sc (row-broadcast), B_f likewise. float4 stream, non-temporal.
__global__ __launch_bounds__(256) void bmat_k(
    const float* __restrict__ sc, const float* __restrict__ fc,
    const float* __restrict__ gsc, const float* __restrict__ gfc,
    float* __restrict__ Bs, float* __restrict__ Bf) {
  const int q = S2048 / 4;                        // 512 float4 per row
  const int per = S2048 * q;                      // 1048576 float4 per matrix
  int t = blockIdx.x * 256 + threadIdx.x;         // grid covers 2*per
  const float* src; const float* g; float* dst; int idx;
  if (t < per) { src = sc; g = gsc; dst = Bs; idx = t; }
  else         { src = fc; g = gfc; dst = Bf; idx = t - per; }
  v4f a  = __builtin_nontemporal_load((const v4f*)src + idx);
  v4f gv = ((const v4f*)g)[idx & (q - 1)];
  __builtin_nontemporal_store(a * gv, (v4f*)dst + idx);
}

__device__ __forceinline__ float block_reduce_sum(float v, float* red) {
  int tid = threadIdx.x;
  red[tid] = v;
  __syncthreads();
  for (int s = 128; s > 0; s >>= 1) {
    if (tid < s) red[tid] += red[tid + s];
    __syncthreads();
  }
  float r = red[0];
  __syncthreads();
  return r;
}

// O = layernorm(A + B) * g + be  (blockIdx.x selects sc/fc stream), n = 2048
__global__ __launch_bounds__(256) void add_ln_k(
    const float* __restrict__ a0, const float* __restrict__ b0, float* __restrict__ o0,
    const float* __restrict__ a1, const float* __restrict__ b1, float* __restrict__ o1,
    const float* __restrict__ g, const float* __restrict__ be) {
  __shared__ float red[256];
  const float* A = blockIdx.x ? a1 : a0;
  const float* B = blockIdx.x ? b1 : b0;
  float* O = blockIdx.x ? o1 : o0;
  const int tid = threadIdx.x;
  float v[8]; float s = 0.f;
  for (int r = 0; r < 8; ++r) { int i = tid + 256 * r; v[r] = A[i] + B[i]; s += v[r]; }
  float mu = block_reduce_sum(s, red) * (1.f / 2048.f);
  float s2 = 0.f;
  for (int r = 0; r < 8; ++r) { float d = v[r] - mu; s2 += d * d; }
  float var = block_reduce_sum(s2, red) * (1.f / 2048.f);
  float rs = rsqrtf(var + 1e-5f);
  for (int r = 0; r < 8; ++r) { int i = tid + 256 * r; O[i] = (v[r] - mu) * rs * g[i] + be[i]; }
}

__global__ __launch_bounds__(256) void prod_k(const float* __restrict__ a,
                                              const float* __restrict__ b,
                                              float* __restrict__ o) {
  int i = blockIdx.x * 256 + threadIdx.x;         // grid covers 2048
  o[i] = a[i] * b[i];
}

// logits = a1 @ W_aw2 + b (2048 -> 3), then softmax into w3[3]
__global__ __launch_bounds__(256) void aw2_softmax_k(
    const float* __restrict__ W, const float* __restrict__ b,
    const float* __restrict__ x, float* __restrict__ w3) {
  __shared__ float red[256];
  float s0 = 0.f, s1 = 0.f, s2 = 0.f;
  for (int i = threadIdx.x; i < 2048; i += 256) {
    float xv = x[i];
    s0 = fmaf(xv, W[i * 3 + 0], s0);
    s1 = fmaf(xv, W[i * 3 + 1], s1);
    s2 = fmaf(xv, W[i * 3 + 2], s2);
  }
  float t0 = block_reduce_sum(s0, red);
  float t1 = block_reduce_sum(s1, red);
  float t2 = block_reduce_sum(s2, red);
  if (threadIdx.x == 0) {
    float l0 = t0 + b[0], l1 = t1 + b[1], l2 = t2 + b[2];
    float m = fmaxf(l0, fmaxf(l1, l2));
    float e0 = expf(l0 - m), e1 = expf(l1 - m), e2 = expf(l2 - m);
    float inv = 1.f / (e0 + e1 + e2);
    w3[0] = e0 * inv; w3[1] = e1 * inv; w3[2] = e2 * inv;
  }
}

__global__ __launch_bounds__(256) void final_k(
    const float* __restrict__ osc, const float* __restrict__ ofc,
    const float* __restrict__ fused, const float* __restrict__ w3,
    const float* __restrict__ x, const float* __restrict__ alpha,
    float* __restrict__ out) {
  int i = blockIdx.x * 256 + threadIdx.x;         // grid covers 2048
  float sa = sigmf(alpha[0]);
  out[i] = w3[0] * osc[i] + w3[1] * ofc[i] + w3[2] * fused[i] + sa * x[i];
}

// ---------------------------------------------------------------------------
// Host-side orchestration
// ---------------------------------------------------------------------------
extern "C" void kernel_launch(void* const* d_in, const int* in_sizes, int n_in,
                              void* d_out, int out_size, void* d_ws, size_t ws_size,
                              hipStream_t stream) {
  (void)in_sizes; (void)n_in; (void)out_size; (void)ws_size;
  // Inputs in setup_inputs() dict order (leaves, insertion order):
  const float* x        = (const float*)d_in[0];
  const float* sc       = (const float*)d_in[1];
  const float* fc       = (const float*)d_in[2];
  const float* Wsc_in   = (const float*)d_in[3];
  const float* bsc_in   = (const float*)d_in[4];
  // d_in[5] = cell_sc.g.W  -- unused (hs == 0)
  const float* bsc_g    = (const float*)d_in[6];
  const float* Wsc_out  = (const float*)d_in[7];
  const float* bsc_out  = (const float*)d_in[8];
  const float* Wfc_in   = (const float*)d_in[9];
  const float* bfc_in   = (const float*)d_in[10];
  // d_in[11] = cell_fc.g.W -- unused
  const float* bfc_g    = (const float*)d_in[12];
  const float* Wfc_out  = (const float*)d_in[13];
  const float* bfc_out  = (const float*)d_in[14];
  // d_in[15..22]: sc_gate/fc_gate/tr_sc/tr_fc -- do not affect outputs
  const float* nw       = (const float*)d_in[23];
  const float* Wcpl     = (const float*)d_in[24];
  const float* bcpl     = (const float*)d_in[25];
  const float* temp     = (const float*)d_in[26];
  const float* Wav      = (const float*)d_in[27];
  const float* bav      = (const float*)d_in[28];
  const float* Wao      = (const float*)d_in[29];
  const float* bao      = (const float*)d_in[30];
  const float* ln_g     = (const float*)d_in[31];
  const float* ln_b     = (const float*)d_in[32];
  const float* Wf1      = (const float*)d_in[33];
  const float* bf1      = (const float*)d_in[34];
  const float* Wf2      = (const float*)d_in[35];
  const float* bf2      = (const float*)d_in[36];
  const float* Waw1     = (const float*)d_in[37];
  const float* baw1     = (const float*)d_in[38];
  const float* Waw2     = (const float*)d_in[39];
  const float* baw2     = (const float*)d_in[40];
  const float* alpha    = (const float*)d_in[41];

  // Output layout: (output, usc, ufc, B_s, B_f, lam) flat
  float* o       = (float*)d_out;
  float* o_usc   = o + 2048;
  float* o_ufc   = o + 4096;
  float* o_Bs    = o + 6144;
  float* o_Bf    = o + 6144 + 4194304;
  float* o_lam   = o + 6144 + 2 * 4194304;

  // Workspace (floats). hsr_sc|hsr_fc adjacent -> free concat for coupling;
  // out_sc|out_fc adjacent -> free concat for aw1.
  float* ws      = (float*)d_ws;
  float* hsr_sc  = ws + 0;       // 2048
  float* hsr_fc  = ws + 2048;    // 2048
  float* out_sc  = ws + 4096;    // 2048
  float* out_fc  = ws + 6144;    // 2048
  float* v_sc    = ws + 8192;    // 2048
  float* v_fc    = ws + 10240;   // 2048
  float* sc_attn = ws + 12288;   // 2048
  float* fc_attn = ws + 14336;   // 2048
  float* cat3    = ws + 16384;   // 6144: [sc_o | fc_o | sc_o*fc_o]
  float* h1      = ws + 22528;   // 4096
  float* fused   = ws + 26624;   // 2048
  float* a1      = ws + 28672;   // 2048
  float* w3      = ws + 30720;   // 4 (3 used)
  float* g_sc    = ws + 30724;   // 2048
  float* g_fc    = ws + 32772;   // 2048
  float* part0   = ws + 36864;   // KSPLIT*4096 = 131072
  float* part1   = ws + 167936;  // KSPLIT*2048 = 65536  (total ~934 KB)

  const dim3 g16(16, KSPLIT), g32(32, KSPLIT);    // fout 2048 / 4096

  // 1) gate vectors g = sigmoid(b_g) (hs==0 kills the matvec)
  gprep_k<<<8, 256, 0, stream>>>(bsc_g, bfc_g, g_sc, g_fc);
  // 2) gated adjacencies B_s, B_f (the two 16MB outputs)
  bmat_k<<<8192, 256, 0, stream>>>(sc, fc, g_sc, g_fc, o_Bs, o_Bf);
  // 3) u = tanh(x@W_in + b); hsr = tanh(u)
  gemv_part_k<2048><<<g16, 256, 0, stream>>>(Wsc_in, x, part0, 2048);
  reduce_act_k<ACT_TANH, true><<<8, 256, 0, stream>>>(part0, bsc_in, o_usc, hsr_sc, 2048, nullptr, nullptr);
  gemv_part_k<2048><<<g16, 256, 0, stream>>>(Wfc_in, x, part0, 2048);
  reduce_act_k<ACT_TANH, true><<<8, 256, 0, stream>>>(part0, bfc_in, o_ufc, hsr_fc, 2048, nullptr, nullptr);
  // 4) out = hsr @ W_out + b
  gemv_part_k<2048><<<g16, 256, 0, stream>>>(Wsc_out, hsr_sc, part0, 2048);
  reduce_act_k<ACT_NONE, false><<<8, 256, 0, stream>>>(part0, bsc_out, out_sc, nullptr, 2048, nullptr, nullptr);
  gemv_part_k<2048><<<g16, 256, 0, stream>>>(Wfc_out, hsr_fc, part0, 2048);
  reduce_act_k<ACT_NONE, false><<<8, 256, 0, stream>>>(part0, bfc_out, out_fc, nullptr, 2048, nullptr, nullptr);
  // 5) lam = sigmoid(sigmoid([hsr_sc,hsr_fc]@Wc + bc) * sigmoid(nw) / T)
  gemv_part_k<2048><<<g16, 256, 0, stream>>>(Wcpl, hsr_sc, part0, 4096);
  reduce_act_k<ACT_LAM, false><<<8, 256, 0, stream>>>(part0, bcpl, o_lam, nullptr, 2048, nw, temp);
  // 6) attention: v = Wv(out_{other}); attn = Wo(v)  (one matrix read, 2 vecs)
  gemv2_part_k<<<g16, 256, 0, stream>>>(Wav, out_fc, out_sc, part0, part1);
  reduce2_k<<<8, 256, 0, stream>>>(part0, part1, bav, v_sc, v_fc);
  gemv2_part_k<<<g16, 256, 0, stream>>>(Wao, v_sc, v_fc, part0, part1);
  reduce2_k<<<8, 256, 0, stream>>>(part0, part1, bao, sc_attn, fc_attn);
  // 7) sc_o/fc_o = LN(out + attn) straight into the fuse-concat buffer
  add_ln_k<<<2, 256, 0, stream>>>(out_sc, sc_attn, cat3,
                                  out_fc, fc_attn, cat3 + 2048, ln_g, ln_b);
  prod_k<<<8, 256, 0, stream>>>(cat3, cat3 + 2048, cat3 + 4096);
  // 8) fused = relu(fuse2(relu(fuse1(cat3))))
  gemv_part_k<4096><<<g32, 256, 0, stream>>>(Wf1, cat3, part0, 6144);
  reduce_act_k<ACT_RELU, false><<<16, 256, 0, stream>>>(part0, bf1, h1, nullptr, 4096, nullptr, nullptr);
  gemv_part_k<2048><<<g16, 256, 0, stream>>>(Wf2, h1, part0, 4096);
  reduce_act_k<ACT_RELU, false><<<8, 256, 0, stream>>>(part0, bf2, fused, nullptr, 2048, nullptr, nullptr);
  // 9) adaptive weights: a1 = relu([out_sc,out_fc]@Waw1 + b); w = softmax(aw2(a1))
  gemv_part_k<2048><<<g16, 256, 0, stream>>>(Waw1, out_sc, part0, 4096);
  reduce_act_k<ACT_RELU, false><<<8, 256, 0, stream>>>(part0, baw1, a1, nullptr, 2048, nullptr, nullptr);
  aw2_softmax_k<<<1, 256, 0, stream>>>(Waw2, baw2, a1, w3);
  // 10) output = w0*out_sc + w1*out_fc + w2*fused + sigmoid(alpha)*x
  final_k<<<8, 256, 0, stream>>>(out_sc, out_fc, fused, w3, x, alpha, o);
}